// ProjectLatLongSphere_31516470018583
// MI455X (gfx1250) — compile-verified
//
#include <hip/hip_runtime.h>

// ProjectLatLongSphere: gather + 4-way bilinear scatter-add + [cell,H] -> [t,H,g] transpose.
// Memory/atomic bound (~4.4 GB moved vs 1.5 GFLOP; AI ~0.35 flop/B): no WMMA applies.
// CDNA5 paths used: no-return global_atomic_add_f32, SMEM-scalarized index loads,
// global_prefetch_b8, and TDM (tensor_load_to_lds + s_wait_tensorcnt) for the transpose.

#define H_DIM 128
#define GRID_CELLS 512          // SPHERE_LAT * SPHERE_LONG = 16*32
#define WAVES_PER_BLOCK 8       // 256 threads, wave32
#define EDGES_PER_WAVE 4

#define TILE_G 32
#define TILE_H 32
#define TILE_PITCH 33           // +1 dword row pad (TDM pad_amount) -> conflict-free columns

typedef unsigned int u32x4 __attribute__((ext_vector_type(4)));
typedef int          i32x8 __attribute__((ext_vector_type(8)));
typedef int          i32x4 __attribute__((ext_vector_type(4)));

__device__ __forceinline__ void atomic_add_f32_noret(float* p, float v) {
    // relaxed, agent scope, result discarded -> global_atomic_add_f32 (no-return, STOREcnt)
    (void)__hip_atomic_fetch_add(p, v, __ATOMIC_RELAXED, __HIP_MEMORY_SCOPE_AGENT);
}

// ---------------------------------------------------------------------------
// Zero-fill (out_size is always a multiple of 4 since H=128)
// ---------------------------------------------------------------------------
__global__ __launch_bounds__(256) void zero_kernel(float4* __restrict__ p, long n4) {
    long i = (long)blockIdx.x * blockDim.x + threadIdx.x;
    if (i < n4) p[i] = make_float4(0.f, 0.f, 0.f, 0.f);
}

// ---------------------------------------------------------------------------
// Path A scatter: accumulate into row-major acc[cell][h] (coalesced atomics).
// One wave per group of EDGES_PER_WAVE edges; lane l owns h = 4l..4l+3.
// ---------------------------------------------------------------------------
__global__ __launch_bounds__(256) void scatter_rows_kernel(
    const float* __restrict__ x,
    const int*   __restrict__ index,   // [4, E]
    const float* __restrict__ delta,   // [4, E]
    const int*   __restrict__ sidx,    // [E]
    float* __restrict__ acc,           // [cells, H]
    int E)
{
    const int lane = threadIdx.x & 31;
    // force SGPR so edge-uniform loads below become scalar (SMEM) loads
    const int wave = __builtin_amdgcn_readfirstlane((int)(threadIdx.x >> 5));
    const long e0 = ((long)blockIdx.x * WAVES_PER_BLOCK + wave) * EDGES_PER_WAVE;
    if (e0 >= E) return;

    // Prefetch this wave's source rows (whole 512B row covered; global_prefetch_b8).
#pragma unroll
    for (int k = 0; k < EDGES_PER_WAVE; ++k) {
        long e = e0 + k;
        if (e < E) {
            int s = sidx[e];  // scalar load
            __builtin_prefetch(&x[(long)s * H_DIM + lane * 4], 0, 1);
        }
    }

#pragma unroll
    for (int k = 0; k < EDGES_PER_WAVE; ++k) {
        long e = e0 + k;
        if (e >= E) break;
        const int s = sidx[e];
        const float4 v = *reinterpret_cast<const float4*>(&x[(long)s * H_DIM + lane * 4]);
#pragma unroll
        for (int i = 0; i < 4; ++i) {
            const int   cell = index[(long)i * E + e];  // scalar load
            const float d    = delta[(long)i * E + e];  // scalar load
            float* p = acc + (long)cell * H_DIM + lane * 4;
            atomic_add_f32_noret(p + 0, v.x * d);
            atomic_add_f32_noret(p + 1, v.y * d);
            atomic_add_f32_noret(p + 2, v.z * d);
            atomic_add_f32_noret(p + 3, v.w * d);
        }
    }
}

// ---------------------------------------------------------------------------
// Path A transpose via TDM: each wave DMAs one [32 g x 32 h] f32 tile of
// acc[row][h] into its private LDS region (hardware +1-dword row padding),
// waits on TENSORcnt, then stores the transposed tile to out[t][h][g].
// No cross-wave barrier needed.
// ---------------------------------------------------------------------------
__global__ __launch_bounds__(256) void transpose_tdm_kernel(
    const float* __restrict__ acc, float* __restrict__ out, int length)
{
    __shared__ float tiles[WAVES_PER_BLOCK][TILE_G][TILE_PITCH];

    const int  lane = threadIdx.x & 31;
    const int  wave = __builtin_amdgcn_readfirstlane((int)(threadIdx.x >> 5));
    const long tilesPerT = (GRID_CELLS / TILE_G) * (H_DIM / TILE_H);   // 16*4 = 64
    const long tileId = (long)blockIdx.x * WAVES_PER_BLOCK + wave;     // nTiles % 8 == 0

    const int t  = (int)(tileId / tilesPerT);
    const int r  = (int)(tileId % tilesPerT);
    const int g0 = (r & 15) << 5;          // 16 g-tiles of 32
    const int h0 = (r >> 4) << 5;          // 4  h-tiles of 32

    const int rows = length * GRID_CELLS;  // tensor_dim1 (for OOB checks)

    // LDS byte offset of this wave's tile (generic->LDS: low 32 bits are the LDS addr)
    const unsigned ldsAddr = (unsigned)(uintptr_t)(&tiles[wave][0][0]);
    const unsigned long long gaddr =
        (unsigned long long)(uintptr_t)(acc + ((long)t * GRID_CELLS + g0) * H_DIM + h0);

    // ---- Tensor DMA descriptor (CDNA5 ISA) ----
    u32x4 d0;
    d0[0] = 1u;                                          // count=1 (valid), user mode
    d0[1] = ldsAddr;                                     // lds_addr
    d0[2] = (unsigned)(gaddr & 0xFFFFFFFFu);             // global_addr[31:0]
    d0[3] = (unsigned)((gaddr >> 32) & 0x1FFFFFFu)       // global_addr[56:32]
          | (2u << 30);                                  // type=2 ("image")

    i32x8 d1;
    d1[0] = (2 << 16)        // data_size = 4 bytes
          | (1 << 20)        // pad_enable (LDS dest padding)
          | (4 << 22);       // pad_interval code 4 -> every 32 dwords; pad_amount code 0 -> 1 dword
    d1[1] = (H_DIM & 0xFFFF) << 16;                      // tensor_dim0[15:0] (=128)
    d1[2] = ((rows & 0xFFFF) << 16);                     // tensor_dim0[31:16]=0 | tensor_dim1[15:0]
    d1[3] = ((rows >> 16) & 0xFFFF)                      // tensor_dim1[31:16]
          | (TILE_H << 16);                              // tile_dim0 = 32 (along h)
    d1[4] = TILE_G;                                      // tile_dim1 = 32 (rows), tile_dim2 = 0
    d1[5] = H_DIM;                                       // tensor_dim0_stride[31:0] = 128
    d1[6] = 0;                                           // stride0[47:32]=0 | stride1[15:0]=0
    d1[7] = (int)(((long)GRID_CELLS * H_DIM) >> 16);     // stride1[47:16] (65536 -> 1)

    i32x4 dz4 = {0, 0, 0, 0};                            // groups 2/3: 2-D tensor
    i32x8 dz8 = {0, 0, 0, 0, 0, 0, 0, 0};                // trailing group (6-arg form)

    __builtin_amdgcn_tensor_load_to_lds(d0, d1, dz4, dz4, dz8, 0);
    __builtin_amdgcn_s_wait_tensorcnt(0);                // wave-local DMA completion

    // Store transposed: lane owns g = g0+lane; column reads tile[lane][h] are
    // conflict-free thanks to the 33-dword pitch; stores coalesce along g.
    float* o = out + ((long)t * H_DIM + h0) * GRID_CELLS + g0 + lane;
#pragma unroll
    for (int h = 0; h < TILE_H; ++h) {
        o[(long)h * GRID_CELLS] = tiles[wave][lane][h];
    }
}

// ---------------------------------------------------------------------------
// Path B (fallback, ws too small): scatter straight into transposed output.
// ---------------------------------------------------------------------------
__global__ __launch_bounds__(256) void scatter_transposed_kernel(
    const float* __restrict__ x,
    const int*   __restrict__ index,
    const float* __restrict__ delta,
    const int*   __restrict__ sidx,
    float* __restrict__ out,           // [length, H, 512]
    int E)
{
    const int lane = threadIdx.x & 31;
    const int wave = __builtin_amdgcn_readfirstlane((int)(threadIdx.x >> 5));
    const long e0 = ((long)blockIdx.x * WAVES_PER_BLOCK + wave) * EDGES_PER_WAVE;
    if (e0 >= E) return;

#pragma unroll
    for (int k = 0; k < EDGES_PER_WAVE; ++k) {
        long e = e0 + k;
        if (e < E) {
            int s = sidx[e];
            __builtin_prefetch(&x[(long)s * H_DIM + lane * 4], 0, 1);
        }
    }

#pragma unroll
    for (int k = 0; k < EDGES_PER_WAVE; ++k) {
        long e = e0 + k;
        if (e >= E) break;
        const int s = sidx[e];
        const float4 v = *reinterpret_cast<const float4*>(&x[(long)s * H_DIM + lane * 4]);
#pragma unroll
        for (int i = 0; i < 4; ++i) {
            const int   cell = index[(long)i * E + e];
            const float d    = delta[(long)i * E + e];
            const int   t = cell >> 9;        // / 512
            const int   g = cell & 511;       // % 512
            float* p = out + ((long)t * H_DIM + lane * 4) * GRID_CELLS + g;
            atomic_add_f32_noret(p + 0 * GRID_CELLS, v.x * d);
            atomic_add_f32_noret(p + 1 * GRID_CELLS, v.y * d);
            atomic_add_f32_noret(p + 2 * GRID_CELLS, v.z * d);
            atomic_add_f32_noret(p + 3 * GRID_CELLS, v.w * d);
        }
    }
}

// ---------------------------------------------------------------------------
extern "C" void kernel_launch(void* const* d_in, const int* in_sizes, int n_in,
                              void* d_out, int out_size, void* d_ws, size_t ws_size,
                              hipStream_t stream) {
    const float* x     = (const float*)d_in[0];
    // d_in[1] = length scalar (unused; derived from out_size)
    const int*   index = (const int*)d_in[2];
    const float* delta = (const float*)d_in[3];
    const int*   sidx  = (const int*)d_in[4];
    float*       out   = (float*)d_out;

    const int  E      = in_sizes[4];
    const long nOut   = (long)out_size;                         // length * H * 512
    const int  length = (int)(nOut / ((long)H_DIM * GRID_CELLS));

    const int edgesPerBlock = WAVES_PER_BLOCK * EDGES_PER_WAVE;
    const int scatterBlocks = (E + edgesPerBlock - 1) / edgesPerBlock;
    const int zeroBlocks    = (int)((nOut / 4 + 255) / 256);

    const size_t accBytes = (size_t)nOut * sizeof(float);

    if (ws_size >= accBytes) {
        // Path A: coalesced row-major accumulate in workspace, TDM-fed transpose.
        float* acc = (float*)d_ws;
        zero_kernel<<<zeroBlocks, 256, 0, stream>>>((float4*)acc, nOut / 4);
        scatter_rows_kernel<<<scatterBlocks, 256, 0, stream>>>(x, index, delta, sidx, acc, E);
        const long nTiles = (long)length * (GRID_CELLS / TILE_G) * (H_DIM / TILE_H); // %8 == 0
        transpose_tdm_kernel<<<(int)(nTiles / WAVES_PER_BLOCK), 256, 0, stream>>>(acc, out, length);
    } else {
        // Path B: scatter directly into transposed output.
        zero_kernel<<<zeroBlocks, 256, 0, stream>>>((float4*)out, nOut / 4);
        scatter_transposed_kernel<<<scatterBlocks, 256, 0, stream>>>(x, index, delta, sidx, out, E);
    }
}